// CCELoss_83906481095114
// MI455X (gfx1250) — compile-verified
//
#include <hip/hip_runtime.h>
#include <hip/hip_bf16.h>

typedef float v2f __attribute__((ext_vector_type(2)));
typedef float v8f __attribute__((ext_vector_type(8)));
typedef int   v4i __attribute__((vector_size(16)));

#define NROWS 4096
#define DK    512
#define VCOLS 128000

#define BM 128
#define BN 128
#define BK 16
#define LDA 20   // padded LDS row stride in floats (16B-aligned rows, conflict-free frag reads)

#if __has_builtin(__builtin_amdgcn_global_load_async_to_lds_b128)
#define HAVE_ASYNC_LDS 1
typedef __attribute__((address_space(1))) v4i* gv4i_ptr;   // global int4*
typedef __attribute__((address_space(3))) v4i* lv4i_ptr;   // LDS int4*
#else
#define HAVE_ASYNC_LDS 0
#endif

__device__ __forceinline__ void wait_async0() {
#if HAVE_ASYNC_LDS
#if __has_builtin(__builtin_amdgcn_s_wait_asynccnt)
    __builtin_amdgcn_s_wait_asynccnt(0);
#else
    asm volatile("s_wait_asynccnt 0" ::: "memory");
#endif
#endif
}

// ---------------------------------------------------------------------------
// WMMA compute on one 128x16 x 16x128 LDS slab (per-wave 32x64 patch)
// ---------------------------------------------------------------------------
__device__ __forceinline__ void compute_tile(
    const float* __restrict__ Asb, const float* __restrict__ Bsb,
    v8f (&acc)[2][4], int wm, int wn, int lrow, int khalf)
{
    #pragma unroll
    for (int kk = 0; kk < BK; kk += 4) {
        const int kidx = kk + 2 * khalf;

        v2f afrag[2];
        #pragma unroll
        for (int mt = 0; mt < 2; ++mt) {
            const int r = wm * 32 + mt * 16 + lrow;
            afrag[mt].x = Asb[r * LDA + kidx];
            afrag[mt].y = Asb[r * LDA + kidx + 1];
        }
        v2f bfrag[4];
        #pragma unroll
        for (int nt = 0; nt < 4; ++nt) {
            const int c = wn * 64 + nt * 16 + lrow;
            bfrag[nt].x = Bsb[c * LDA + kidx];
            bfrag[nt].y = Bsb[c * LDA + kidx + 1];
        }

        #pragma unroll
        for (int mt = 0; mt < 2; ++mt)
            #pragma unroll
            for (int nt = 0; nt < 4; ++nt)
                acc[mt][nt] = __builtin_amdgcn_wmma_f32_16x16x4_f32(
                    false, afrag[mt], false, bfrag[nt],
                    (short)0, acc[mt][nt], false, false);
    }
}

// ---------------------------------------------------------------------------
// Kernel 1: logits = E @ W^T + b  (fp32 WMMA, async-LDS double-buffered)
// ---------------------------------------------------------------------------
__global__ __launch_bounds__(256) void gemm_bias_kernel(
    const float* __restrict__ E, const float* __restrict__ W,
    const float* __restrict__ bias, float* __restrict__ out)
{
    __shared__ float As0[BM * LDA];
    __shared__ float As1[BM * LDA];
    __shared__ float Bs0[BN * LDA];
    __shared__ float Bs1[BN * LDA];

    const int tid = threadIdx.x;
    const int bn  = blockIdx.x;   // 0..999  (V / 128)
    const int bm  = blockIdx.y;   // 0..31   (N / 128)

    const int wave  = tid >> 5;   // 0..7 (wave32)
    const int lane  = tid & 31;
    const int lrow  = lane & 15;
    const int khalf = lane >> 4;

    const int wm = wave & 3;      // row offset wm*32
    const int wn = wave >> 2;     // col offset wn*64

    v8f acc[2][4];
    {
        v8f zero = {};
        #pragma unroll
        for (int i = 0; i < 2; ++i)
            #pragma unroll
            for (int j = 0; j < 4; ++j) acc[i][j] = zero;
    }

    const int qcol = (tid & 3) * 4;   // 0,4,8,12
    const int qrow = tid >> 2;        // 0..63

    const float* Ebase = E + (size_t)(bm * BM) * DK;
    const float* Wbase = W + (size_t)(bn * BN) * DK;

#if HAVE_ASYNC_LDS
    // ---- async global->LDS DMA pipeline (ASYNCcnt-tracked, no VGPR staging)
    #define ISSUE_TILE(Asb, Bsb, K0)                                            \
        do {                                                                    \
            _Pragma("unroll")                                                   \
            for (int h = 0; h < 2; ++h) {                                       \
                const int r = qrow + h * 64;                                    \
                __builtin_amdgcn_global_load_async_to_lds_b128(                 \
                    (gv4i_ptr)(Ebase + (size_t)r * DK + (K0) + qcol),           \
                    (lv4i_ptr)((Asb) + r * LDA + qcol), 0, 0);                  \
                __builtin_amdgcn_global_load_async_to_lds_b128(                 \
                    (gv4i_ptr)(Wbase + (size_t)r * DK + (K0) + qcol),           \
                    (lv4i_ptr)((Bsb) + r * LDA + qcol), 0, 0);                  \
            }                                                                   \
        } while (0)

    ISSUE_TILE(As0, Bs0, 0);
    #pragma unroll 1
    for (int k0 = 0; k0 < DK; k0 += 2 * BK) {
        wait_async0();              // own slab-A DMA complete
        __syncthreads();            // everyone's slab-A complete, prev consumers done
        if (k0 + BK < DK) ISSUE_TILE(As1, Bs1, k0 + BK);   // overlaps compute
        compute_tile(As0, Bs0, acc, wm, wn, lrow, khalf);

        wait_async0();
        __syncthreads();
        if (k0 + 2 * BK < DK) ISSUE_TILE(As0, Bs0, k0 + 2 * BK);
        compute_tile(As1, Bs1, acc, wm, wn, lrow, khalf);
    }
    #undef ISSUE_TILE
#else
    // ---- fallback: double-buffered LDS + explicit VGPR prefetch of next slab
    float4 pA[2], pB[2];
    #pragma unroll
    for (int h = 0; h < 2; ++h) {
        const int r = qrow + h * 64;
        pA[h] = *(const float4*)(Ebase + (size_t)r * DK + 0 + qcol);
        pB[h] = *(const float4*)(Wbase + (size_t)r * DK + 0 + qcol);
    }

    #define STORE_TILE(Asb, Bsb)                                                \
        do {                                                                    \
            _Pragma("unroll")                                                   \
            for (int h = 0; h < 2; ++h) {                                       \
                const int r = qrow + h * 64;                                    \
                *(float4*)(&(Asb)[r * LDA + qcol]) = pA[h];                     \
                *(float4*)(&(Bsb)[r * LDA + qcol]) = pB[h];                     \
            }                                                                   \
        } while (0)
    #define LOAD_NEXT(K0)                                                       \
        do {                                                                    \
            _Pragma("unroll")                                                   \
            for (int h = 0; h < 2; ++h) {                                       \
                const int r = qrow + h * 64;                                    \
                pA[h] = *(const float4*)(Ebase + (size_t)r * DK + (K0) + qcol); \
                pB[h] = *(const float4*)(Wbase + (size_t)r * DK + (K0) + qcol); \
            }                                                                   \
        } while (0)

    #pragma unroll 1
    for (int k0 = 0; k0 < DK; k0 += 2 * BK) {
        STORE_TILE(As0, Bs0);
        if (k0 + BK < DK) LOAD_NEXT(k0 + BK);
        __syncthreads();
        compute_tile(As0, Bs0, acc, wm, wn, lrow, khalf);

        STORE_TILE(As1, Bs1);
        if (k0 + 2 * BK < DK) LOAD_NEXT(k0 + 2 * BK);
        __syncthreads();
        compute_tile(As1, Bs1, acc, wm, wn, lrow, khalf);
    }
    #undef STORE_TILE
    #undef LOAD_NEXT
#endif

    // Epilogue: add bias, store. C/D layout: VGPR i -> M=i (lanes 0-15), M=i+8 (lanes 16-31).
    #pragma unroll
    for (int nt = 0; nt < 4; ++nt) {
        const int gcol = bn * BN + wn * 64 + nt * 16 + lrow;
        const float bv = bias[gcol];
        #pragma unroll
        for (int mt = 0; mt < 2; ++mt) {
            const int rbase = bm * BM + wm * 32 + mt * 16 + 8 * khalf;
            #pragma unroll
            for (int i = 0; i < 8; ++i) {
                const size_t idx = (size_t)(rbase + i) * VCOLS + gcol;
                out[idx] = acc[mt][nt][i] + bv;
            }
        }
    }
}

// ---------------------------------------------------------------------------
// Kernel 2: per-row online logsumexp minus target logit
// (scalar loads: logits base is d_out+1, only 4B-aligned; still coalesced)
// ---------------------------------------------------------------------------
__global__ __launch_bounds__(256) void rowlse_kernel(
    const float* __restrict__ logits, const long long* __restrict__ target,
    float* __restrict__ rowloss)
{
    __shared__ float sm[256];
    __shared__ float ss[256];

    const int row = blockIdx.x;
    const float* p = logits + (size_t)row * VCOLS;

    float m = -INFINITY, s = 0.0f;
    for (int v = threadIdx.x; v < VCOLS; v += 256) {
        const float x = p[v];
        if (x > m) { s = s * __expf(m - x) + 1.0f; m = x; }
        else       { s += __expf(x - m); }
    }
    sm[threadIdx.x] = m;
    ss[threadIdx.x] = s;
    __syncthreads();

    for (int off = 128; off > 0; off >>= 1) {
        if (threadIdx.x < off) {
            const float m1 = sm[threadIdx.x],       s1 = ss[threadIdx.x];
            const float m2 = sm[threadIdx.x + off], s2 = ss[threadIdx.x + off];
            const float M  = fmaxf(m1, m2);
            sm[threadIdx.x] = M;
            ss[threadIdx.x] = s1 * __expf(m1 - M) + s2 * __expf(m2 - M);
        }
        __syncthreads();
    }

    if (threadIdx.x == 0) {
        const float logz = sm[0] + __logf(ss[0]);
        const long long t = target[row];
        rowloss[row] = logz - p[t];
    }
}

// ---------------------------------------------------------------------------
// Kernel 3: mean over rows -> loss scalar
// ---------------------------------------------------------------------------
__global__ __launch_bounds__(256) void mean_kernel(
    const float* __restrict__ rowloss, float* __restrict__ out)
{
    __shared__ float sm[256];
    float s = 0.0f;
    for (int i = threadIdx.x; i < NROWS; i += 256) s += rowloss[i];
    sm[threadIdx.x] = s;
    __syncthreads();
    for (int off = 128; off > 0; off >>= 1) {
        if (threadIdx.x < off) sm[threadIdx.x] += sm[threadIdx.x + off];
        __syncthreads();
    }
    if (threadIdx.x == 0) out[0] = sm[0] / (float)NROWS;
}

// ---------------------------------------------------------------------------
extern "C" void kernel_launch(void* const* d_in, const int* in_sizes, int n_in,
                              void* d_out, int out_size, void* d_ws, size_t ws_size,
                              hipStream_t stream) {
    const float*      E   = (const float*)d_in[0];       // [4096, 512]
    const long long*  tgt = (const long long*)d_in[1];   // [4096] int64
    const float*      W   = (const float*)d_in[2];       // [128000, 512]
    const float*      b   = (const float*)d_in[3];       // [128000]

    float* loss    = (float*)d_out;       // d_out[0]
    float* logits  = loss + 1;            // d_out[1 ..] : [4096, 128000]
    float* rowloss = (float*)d_ws;        // 4096 floats scratch

    dim3 grid(VCOLS / BN, NROWS / BM);    // (1000, 32)
    gemm_bias_kernel<<<grid, 256, 0, stream>>>(E, W, b, logits);
    rowlse_kernel<<<NROWS, 256, 0, stream>>>(logits, tgt, rowloss);
    mean_kernel<<<1, 256, 0, stream>>>(rowloss, loss);
}